// FactorizationMachineOneHot_8778913153593
// MI455X (gfx1250) — compile-verified
//
#include <hip/hip_runtime.h>

typedef float v2f __attribute__((ext_vector_type(2)));
typedef float v4f __attribute__((ext_vector_type(4)));
typedef float v8f __attribute__((ext_vector_type(8)));

#define WAVES    8
#define BLOCK    (WAVES * 32)
#define TILE_M   16
#define GROUP    8                      // K=4 chunks per pipeline stage (K=32)

// ---------------------------------------------------------------------------
// Prep pass over feature pairs: comb[4p..4p+3] = { fc_w[2p], fc_w[2p+1],
//   s[2p], s[2p+1] } with s[f] = sum_d w[f,d]^2.  Collapses the X^2 @ w^2
// GEMM into a matvec (the D-sum is linear), halving fp32 WMMA work.
// ---------------------------------------------------------------------------
__global__ __launch_bounds__(256) void fm_prep(
    const float* __restrict__ fc_w,
    const float* __restrict__ w,
    float* __restrict__ comb,
    int F)
{
    int p  = blockIdx.x * blockDim.x + threadIdx.x;   // feature pair index
    int f0 = 2 * p;
    if (f0 < F) {
        const v4f* wr0 = (const v4f*)(w + (size_t)f0 * 16);
        float s0 = 0.0f, s1 = 0.0f, fc1 = 0.0f;
        #pragma unroll
        for (int i = 0; i < 4; ++i) {
            v4f q = wr0[i];
            s0 += q.x * q.x + q.y * q.y + q.z * q.z + q.w * q.w;
        }
        if (f0 + 1 < F) {
            const v4f* wr1 = (const v4f*)(w + (size_t)(f0 + 1) * 16);
            #pragma unroll
            for (int i = 0; i < 4; ++i) {
                v4f q = wr1[i];
                s1 += q.x * q.x + q.y * q.y + q.z * q.z + q.w * q.w;
            }
            fc1 = fc_w[f0 + 1];
        }
        v4f o; o.x = fc_w[f0]; o.y = fc1; o.z = s0; o.w = s1;
        *(v4f*)(comb + 4 * (size_t)p) = o;
    }
}

// ---------------------------------------------------------------------------
// Main kernel. Double-buffered pipeline: while group g computes, the 32 loads
// of group g+1 are in flight (distinct staging registers -> no WAR stalls).
// Per K=4 chunk: 1x b64 NT (X) + 2x b32 (w) + 1x b128 (comb);
//                1x v_wmma_f32_16x16x4_f32 + 1 pk_mul + 2 pk_fma.
// ---------------------------------------------------------------------------
#define FM_LOADG(S)                                                           \
    {                                                                         \
        __builtin_prefetch(xp + 1024, 0, 0);                                  \
        _Pragma("unroll")                                                     \
        for (int u = 0; u < GROUP; ++u) {                                     \
            a##S[u] = __builtin_nontemporal_load((const v2f*)(xp + 4 * u));   \
            v2f t; t.x = wp[64 * u]; t.y = wp[64 * u + 16];                   \
            b##S[u]  = t;                                                     \
            cs##S[u] = *(const v4f*)(cp + 8 * u);                             \
        }                                                                     \
        xp += 4 * GROUP; cp += 8 * GROUP; wp += 64 * GROUP;                   \
    }

#define FM_COMPG(S)                                                           \
    {                                                                         \
        _Pragma("unroll")                                                     \
        for (int u = 0; u < GROUP; ++u) {                                     \
            acc = __builtin_amdgcn_wmma_f32_16x16x4_f32(                      \
                      false, a##S[u], false, b##S[u], (short)0,               \
                      acc, false, false);                                     \
            v2f fcp = __builtin_shufflevector(cs##S[u], cs##S[u], 0, 1);      \
            v2f spr = __builtin_shufflevector(cs##S[u], cs##S[u], 2, 3);      \
            v2f a2  = a##S[u] * a##S[u];                                      \
            linv = a##S[u] * fcp + linv;                                      \
            ssqv = a2 * spr + ssqv;                                           \
        }                                                                     \
    }

#define FM_CHUNK1()                                                           \
    {                                                                         \
        v2f a    = __builtin_nontemporal_load((const v2f*)xp);                \
        float b0 = wp[0];                                                     \
        float b1 = wp[16];                                                    \
        v4f cs   = *(const v4f*)cp;                                           \
        v2f b; b.x = b0; b.y = b1;                                            \
        acc = __builtin_amdgcn_wmma_f32_16x16x4_f32(                          \
                  false, a, false, b, (short)0, acc, false, false);           \
        v2f fcp = __builtin_shufflevector(cs, cs, 0, 1);                      \
        v2f spr = __builtin_shufflevector(cs, cs, 2, 3);                      \
        v2f a2  = a * a;                                                      \
        linv = a  * fcp + linv;                                               \
        ssqv = a2 * spr + ssqv;                                               \
        xp += 4; cp += 8; wp += 64;                                           \
    }

__global__ __launch_bounds__(BLOCK) void fm_wmma_s(
    const float* __restrict__ X,
    const float* __restrict__ comb,
    const float* __restrict__ fc_b,
    const float* __restrict__ w,
    float* __restrict__ out,
    int F)
{
    __shared__ float sC[WAVES][TILE_M][TILE_M];
    __shared__ float sLin[WAVES][TILE_M];
    __shared__ float sSsq[WAVES][TILE_M];

    const int lane = threadIdx.x & 31;
    const int wave = threadIdx.x >> 5;
    const int m    = lane & 15;
    const int half = lane >> 4;
    const int tile = blockIdx.x;
    const int row  = tile * TILE_M + m;

    const int chunks      = F >> 2;
    const int chunksPerWv = (chunks + WAVES - 1) / WAVES;
    const int c0 = wave * chunksPerWv;
    const int c1 = (c0 + chunksPerWv < chunks) ? (c0 + chunksPerWv) : chunks;
    const int kBeg = __builtin_amdgcn_readfirstlane(c0 << 2);
    const int kEnd = __builtin_amdgcn_readfirstlane(c1 << 2);

    const float* xp = X    + (size_t)row * F + 2 * half + kBeg;
    const float* wp = w    + 2 * half * 16 + m + (size_t)kBeg * 16;
    const float* cp = comb + 4 * half + 2 * (size_t)kBeg;

    v8f acc = {};
    v2f linv = {}, ssqv = {};

    // Pipeline staging buffers (kept in registers; constant indices only).
    v2f aA[GROUP], bA[GROUP]; v4f csA[GROUP];
    v2f aB[GROUP], bB[GROUP]; v4f csB[GROUP];

    const int nGroups = (kEnd - kBeg) >> 5;
    if (nGroups > 0) {
        FM_LOADG(A);
        int rem = nGroups - 1;
        while (rem >= 2) {
            FM_LOADG(B);
            FM_COMPG(A);
            FM_LOADG(A);
            FM_COMPG(B);
            rem -= 2;
        }
        if (rem == 1) {
            FM_LOADG(B);
            FM_COMPG(A);
            FM_COMPG(B);
        } else {
            FM_COMPG(A);
        }
    }
    // <= 7 leftover K=4 chunks
    for (int k = kBeg + (nGroups << 5); k < kEnd; k += 4)
        FM_CHUNK1();

    #pragma unroll
    for (int j = 0; j < 8; ++j)
        sC[wave][half * 8 + j][m] = acc[j];

    float lin = linv.x + linv.y;
    float ssq = ssqv.x + ssqv.y;
    lin += __shfl_xor(lin, 16, 32);
    ssq += __shfl_xor(ssq, 16, 32);
    if (lane < 16) { sLin[wave][m] = lin; sSsq[wave][m] = ssq; }

    __syncthreads();

    if (threadIdx.x < TILE_M) {
        const int r = threadIdx.x;
        float sq = 0.0f, ssqT = 0.0f, linT = 0.0f;
        #pragma unroll
        for (int d = 0; d < TILE_M; ++d) {
            float v1 = 0.0f;
            #pragma unroll
            for (int wv = 0; wv < WAVES; ++wv) v1 += sC[wv][r][d];
            sq += v1 * v1;
        }
        #pragma unroll
        for (int wv = 0; wv < WAVES; ++wv) { linT += sLin[wv][r]; ssqT += sSsq[wv][r]; }

        float z = linT + fc_b[0] + 0.5f * (sq - ssqT);
        out[tile * TILE_M + r] = 1.0f / (1.0f + __expf(-z));
    }
}

// ---------------------------------------------------------------------------
// Fallback (no scratch): two-WMMA version, self-contained.
// ---------------------------------------------------------------------------
#define FM_CHUNK2(kk)                                                         \
    {                                                                         \
        v2f a    = __builtin_nontemporal_load((const v2f*)(Xrow + (kk)));     \
        float b0 = Wbase[(size_t)(kk) * 16];                                  \
        float b1 = Wbase[(size_t)(kk) * 16 + 16];                             \
        v2f c    = *(const v2f*)(Cbase + (kk));                               \
        v2f b;  b.x  = b0;        b.y  = b1;                                  \
        v2f a2; a2.x = a.x * a.x; a2.y = a.y * a.y;                           \
        v2f b2; b2.x = b0 * b0;   b2.y = b1 * b1;                             \
        acc1 = __builtin_amdgcn_wmma_f32_16x16x4_f32(                         \
                   false, a,  false, b,  (short)0, acc1, false, false);       \
        acc2 = __builtin_amdgcn_wmma_f32_16x16x4_f32(                         \
                   false, a2, false, b2, (short)0, acc2, false, false);       \
        lin = fmaf(a.x, c.x, lin);                                            \
        lin = fmaf(a.y, c.y, lin);                                            \
    }

__global__ __launch_bounds__(BLOCK) void fm_wmma_fb(
    const float* __restrict__ X,
    const float* __restrict__ fc_w,
    const float* __restrict__ fc_b,
    const float* __restrict__ w,
    float* __restrict__ out,
    int F)
{
    __shared__ float sC1[WAVES][TILE_M][TILE_M];
    __shared__ float sC2[WAVES][TILE_M][TILE_M];
    __shared__ float sLin[WAVES][TILE_M];

    const int lane = threadIdx.x & 31;
    const int wave = threadIdx.x >> 5;
    const int m    = lane & 15;
    const int half = lane >> 4;
    const int tile = blockIdx.x;
    const int row  = tile * TILE_M + m;

    const int chunks      = F >> 2;
    const int chunksPerWv = (chunks + WAVES - 1) / WAVES;
    const int c0 = wave * chunksPerWv;
    const int c1 = (c0 + chunksPerWv < chunks) ? (c0 + chunksPerWv) : chunks;
    const int kBeg = __builtin_amdgcn_readfirstlane(c0 << 2);
    const int kEnd = __builtin_amdgcn_readfirstlane(c1 << 2);

    const float* __restrict__ Xrow  = X    + (size_t)row * F + 2 * half;
    const float* __restrict__ Wbase = w    + 2 * half * 16 + m;
    const float* __restrict__ Cbase = fc_w + 2 * half;

    v8f acc1 = {}, acc2 = {};
    float lin = 0.0f;

    int k = kBeg;
    const int nGroups = (kEnd - kBeg) >> 4;
    for (int g = 0; g < nGroups; ++g, k += 16) {
        __builtin_prefetch(Xrow + k + 1024, 0, 0);
        FM_CHUNK2(k);
        FM_CHUNK2(k + 4);
        FM_CHUNK2(k + 8);
        FM_CHUNK2(k + 12);
    }
    for (; k < kEnd; k += 4)
        FM_CHUNK2(k);

    #pragma unroll
    for (int j = 0; j < 8; ++j) {
        sC1[wave][half * 8 + j][m] = acc1[j];
        sC2[wave][half * 8 + j][m] = acc2[j];
    }
    lin += __shfl_xor(lin, 16, 32);
    if (lane < 16) sLin[wave][m] = lin;

    __syncthreads();

    if (threadIdx.x < TILE_M) {
        const int r = threadIdx.x;
        float sq = 0.0f, ssq = 0.0f, linT = 0.0f;
        #pragma unroll
        for (int d = 0; d < TILE_M; ++d) {
            float v1 = 0.0f, v2 = 0.0f;
            #pragma unroll
            for (int wv = 0; wv < WAVES; ++wv) { v1 += sC1[wv][r][d]; v2 += sC2[wv][r][d]; }
            sq  += v1 * v1;
            ssq += v2;
        }
        #pragma unroll
        for (int wv = 0; wv < WAVES; ++wv) linT += sLin[wv][r];

        float z = linT + fc_b[0] + 0.5f * (sq - ssq);
        out[tile * TILE_M + r] = 1.0f / (1.0f + __expf(-z));
    }
}

extern "C" void kernel_launch(void* const* d_in, const int* in_sizes, int n_in,
                              void* d_out, int out_size, void* d_ws, size_t ws_size,
                              hipStream_t stream) {
    const float* X    = (const float*)d_in[0];
    const float* fc_w = (const float*)d_in[1];
    const float* fc_b = (const float*)d_in[2];
    const float* w    = (const float*)d_in[3];
    float* out = (float*)d_out;

    const int F = in_sizes[1];      // 50000
    const int B = out_size;         // 4096

    if (ws_size >= (size_t)2 * F * sizeof(float)) {
        float* comb = (float*)d_ws;
        const int nPairs = (F + 1) / 2;
        fm_prep<<<dim3((nPairs + 255) / 256), dim3(256), 0, stream>>>(fc_w, w, comb, F);
        fm_wmma_s<<<dim3(B / TILE_M), dim3(BLOCK), 0, stream>>>(
            X, comb, fc_b, w, out, F);
    } else {
        fm_wmma_fb<<<dim3(B / TILE_M), dim3(BLOCK), 0, stream>>>(
            X, fc_w, fc_b, w, out, F);
    }
}